// Aggregator_62706522522365
// MI455X (gfx1250) — compile-verified
//
#include <hip/hip_runtime.h>
#include <hip/hip_bf16.h>

typedef __attribute__((ext_vector_type(2))) float v2f;
typedef __attribute__((ext_vector_type(8))) float v8f;

#define NUM_KP   512
#define S_PTS    64
#define RAYS     2048          // T * Ro = 2 * 1024
#define BATCH    4
#define MAXSH    48
#define R2       0.0144f       // R^2 = 0.12^2
#define ROWSTR   133           // odd LDS row stride (conflict-free)

// output region offsets (flat float elements)
#define OFF_IDX   ((size_t)0)
#define OFF_SP    ((size_t)4194304)                 // 4*2048*64*8
#define OFF_DST   ((size_t)5767168)                 // + 4*2048*64*3
#define OFF_MSK   ((size_t)9961472)                 // + 4*2048*64*8

// conditional swap keyed on distance, carries index
#define CSW(a,b,ia,ib) do { if ((b) < (a)) {                       \
      float _td=(a); (a)=(b); (b)=_td;                             \
      int   _ti=(ia); (ia)=(ib); (ib)=_ti; } } while(0)

// insert candidate (dv,jv) into ascending 8-list t0..t7 / i0..i7
#define INSERT8(dv,jv) do { float _d=(dv);                         \
    if (_d < t7) { t7=_d; i7=(jv);                                 \
      CSW(t6,t7,i6,i7); CSW(t5,t6,i5,i6); CSW(t4,t5,i4,i5);        \
      CSW(t3,t4,i3,i4); CSW(t2,t3,i2,i3); CSW(t1,t2,i1,i2);        \
      CSW(t0,t1,i0,i1); } } while(0)

__launch_bounds__(128)
__global__ void knn_query_kernel(const float* __restrict__ x,
                                 const float* __restrict__ kp,
                                 float* __restrict__ out)
{
    __shared__ float    kp4[NUM_KP * 4];      // (kx,ky,kz,0) per keypoint
    __shared__ float    b2s[NUM_KP];          // |kp|^2
    __shared__ float    xs[S_PTS * 4];        // (x,y,z,|x|^2) per point
    __shared__ float    d2buf[4][16 * ROWSTR];// per-wave 16 rows x 128-col chunk
    __shared__ unsigned vmask[4];

    const int tid  = threadIdx.x;
    const int wave = tid >> 5;
    const int lane = tid & 31;
    const int nl   = lane & 15;
    const int h    = lane >> 4;

    const int blk = blockIdx.x;               // b*RAYS + ray
    const int b   = blk >> 11;                // RAYS == 2048

    // ---------------- stage 0: cooperative loads ----------------
    for (int i = tid; i < NUM_KP; i += 128) {
        const float kx = kp[(b * NUM_KP + i) * 3 + 0];
        const float ky = kp[(b * NUM_KP + i) * 3 + 1];
        const float kz = kp[(b * NUM_KP + i) * 3 + 2];
        kp4[i * 4 + 0] = kx;  kp4[i * 4 + 1] = ky;
        kp4[i * 4 + 2] = kz;  kp4[i * 4 + 3] = 0.0f;
        b2s[i] = kx * kx + ky * ky + kz * kz;
    }
    if (tid < S_PTS) {
        const size_t pb = ((size_t)blk * S_PTS + tid) * 3;
        const float px = x[pb + 0], py = x[pb + 1], pz = x[pb + 2];
        xs[tid * 4 + 0] = px;  xs[tid * 4 + 1] = py;  xs[tid * 4 + 2] = pz;
        xs[tid * 4 + 3] = px * px + py * py + pz * pz;
    }
    __syncthreads();

    // ---- A fragment (16x4 f32): lanes 0-15 hold K=0,1; lanes 16-31 K=2,3 ----
    const int qb = wave * 16;
    v2f amat;
    amat.x = xs[(qb + nl) * 4 + 2 * h];            // x or z
    amat.y = h ? 0.0f : xs[(qb + nl) * 4 + 1];     // y or pad
    float a2r[8];
#pragma unroll
    for (int r = 0; r < 8; ++r)                    // |x|^2 for rows r+8h
        a2r[r] = xs[(qb + 8 * h + r) * 4 + 3];

    // top-8 ascending list, R^2 sentinels (anything >= R^2 never enters)
    float t0=R2,t1=R2,t2=R2,t3=R2,t4=R2,t5=R2,t6=R2,t7=R2;
    int   i0=-1,i1=-1,i2=-1,i3=-1,i4=-1,i5=-1,i6=-1,i7=-1;

    float* wbuf = &d2buf[wave][0];

    // ---------------- main loop: 4 chunks of 128 keypoints ----------------
    for (int chunk = 0; chunk < 4; ++chunk) {
#pragma unroll
        for (int tt = 0; tt < 8; ++tt) {
            const int n = (chunk * 8 + tt) * 16 + nl;
            v2f bmat;                                  // B 4x16: same half-split
            bmat.x = kp4[n * 4 + 2 * h];
            bmat.y = kp4[n * 4 + 2 * h + 1];           // kp4[n*4+3]==0 for h==1
            const float bn2 = b2s[n];
            v8f acc = {};
            acc = __builtin_amdgcn_wmma_f32_16x16x4_f32(
                false, amat, false, bmat, (short)0, acc, false, false);
#pragma unroll
            for (int r = 0; r < 8; ++r) {              // d2 = a2 + b2 - 2*dot
                const float d2v = __builtin_fmaf(-2.0f, acc[r], a2r[r] + bn2);
                wbuf[(8 * h + r) * ROWSTR + tt * 16 + nl] = d2v;
            }
        }
        // wave-internal LDS RAW fence (cross-lane through d2buf)
        asm volatile("s_wait_dscnt 0" ::: "memory");

        // lanes (p,h=0) scan cols 0..63, (p,h=1) scan cols 64..127 of row p
        const float* row = &wbuf[nl * ROWSTR + h * 64];
        const int cb = chunk * 128 + h * 64;
#pragma unroll 4
        for (int j = 0; j < 64; ++j) {
            const float d2v = row[j];
            INSERT8(d2v, cb + j);
        }
    }

    // ---------------- merge half-wave partial lists ----------------
    const int src = nl + 16;   // snapshot other half's list BEFORE mutating
    const float od0=__shfl(t0,src,32), od1=__shfl(t1,src,32),
                od2=__shfl(t2,src,32), od3=__shfl(t3,src,32),
                od4=__shfl(t4,src,32), od5=__shfl(t5,src,32),
                od6=__shfl(t6,src,32), od7=__shfl(t7,src,32);
    const int   oi0=__shfl(i0,src,32), oi1=__shfl(i1,src,32),
                oi2=__shfl(i2,src,32), oi3=__shfl(i3,src,32),
                oi4=__shfl(i4,src,32), oi5=__shfl(i5,src,32),
                oi6=__shfl(i6,src,32), oi7=__shfl(i7,src,32);
    if (lane < 16) {
        INSERT8(od0,oi0); INSERT8(od1,oi1); INSERT8(od2,oi2); INSERT8(od3,oi3);
        INSERT8(od4,oi4); INSERT8(od5,oi5); INSERT8(od6,oi6); INSERT8(od7,oi7);
    }

    // ---------------- per-ray capped cumsum via ballots ----------------
    const bool valid = (lane < 16) && (i0 >= 0);
    const unsigned bal = (unsigned)__ballot(valid);
    if (lane == 0) vmask[wave] = bal;
    __syncthreads();

    const unsigned m0 = vmask[0], m1 = vmask[1], m2 = vmask[2], m3 = vmask[3];
    const int total = __popc(m0) + __popc(m1) + __popc(m2) + __popc(m3);
    const int num_valid = total < MAXSH ? total : MAXSH;

    if (lane < 16) {
        int below = 0;
        if (wave > 0) below += __popc(m0);
        if (wave > 1) below += __popc(m1);
        if (wave > 2) below += __popc(m2);
        const unsigned own = vmask[wave] & ((2u << lane) - 1);
        const int  csum = below + __popc(own);                // inclusive
        const bool fv   = (i0 >= 0) && (csum <= MAXSH);

        const int    q   = qb + lane;
        const size_t pt  = (size_t)blk * S_PTS + q;
        const int    ofs = b * NUM_KP;

        float* oidx = out + OFF_IDX + pt * 8;
        float* odst = out + OFF_DST + pt * 8;
#define EMIT(k, tk, ik) do { const bool vs = fv && ((ik) >= 0);              \
        oidx[k] = vs ? (float)((ik) + ofs) : -1.0f;                          \
        odst[k] = vs ? sqrtf(fmaxf((tk), 1e-12f)) : 0.0f; } while(0)
        EMIT(0,t0,i0); EMIT(1,t1,i1); EMIT(2,t2,i2); EMIT(3,t3,i3);
        EMIT(4,t4,i4); EMIT(5,t5,i5); EMIT(6,t6,i6); EMIT(7,t7,i7);
#undef EMIT

        float* osp = out + OFF_SP + pt * 3;
        osp[0] = fv ? xs[q * 4 + 0] : 0.0f;
        osp[1] = fv ? xs[q * 4 + 1] : 0.0f;
        osp[2] = fv ? xs[q * 4 + 2] : 0.0f;
    }

    if (tid < MAXSH)
        out[OFF_MSK + (size_t)blk * MAXSH + tid] = (tid < num_valid) ? 1.0f : 0.0f;
}

extern "C" void kernel_launch(void* const* d_in, const int* in_sizes, int n_in,
                              void* d_out, int out_size, void* d_ws, size_t ws_size,
                              hipStream_t stream) {
    const float* x  = (const float*)d_in[0];   // [4,2,1024,64,3]
    const float* kp = (const float*)d_in[1];   // [4,512,3]
    float* out = (float*)d_out;
    (void)in_sizes; (void)n_in; (void)out_size; (void)d_ws; (void)ws_size;

    dim3 grid(BATCH * RAYS);   // 8192 blocks: one ray (64 points) each
    dim3 block(128);           // 4 wave32, 16 points per wave
    hipLaunchKernelGGL(knn_query_kernel, grid, block, 0, stream, x, kp, out);
}